// HSTv7_1Ultimate_84963043049706
// MI455X (gfx1250) — compile-verified
//
#include <hip/hip_runtime.h>
#include <math.h>

// ---- problem constants (from reference) ----
#define SS   2048      // sequence length
#define DD   1024      // d_model
#define HH   16        // heads
#define HD   64        // head dim
#define NBLK 32        // number of 64-token blocks (S / 64)
#define RH   128       // router hidden
#define NEG_BIG (-1.0e9f)

typedef __attribute__((ext_vector_type(2))) float v2f;
typedef __attribute__((ext_vector_type(4))) float v4f;
typedef __attribute__((ext_vector_type(8))) float v8f;

// D = A(16x4) * B(4x16) + C, full fp32 (exact vs reference)
__device__ __forceinline__ v8f wmma4(v2f a, v2f b, v8f c) {
  return __builtin_amdgcn_wmma_f32_16x16x4_f32(
      /*neg_a=*/false, a, /*neg_b=*/false, b,
      /*c_mod=*/(short)0, c, /*reuse_a=*/false, /*reuse_b=*/false);
}

// ---- CDNA5 async global->LDS path (guarded; falls back to VGPR staging) ----
#if defined(__has_builtin)
#if __has_builtin(__builtin_amdgcn_global_load_async_to_lds_b128) && \
    __has_builtin(__builtin_amdgcn_s_wait_asynccnt)
#define USE_ASYNC_LDS 1
#endif
#endif

#ifdef USE_ASYNC_LDS
// builtin signature (from hipcc diagnostic): arg0 is
//   'int __attribute__((vector_size(16))) __device__ *'  (= int4 addrspace(1)*)
typedef int g4i __attribute__((vector_size(16)));
typedef g4i __attribute__((address_space(1))) glb4i;
typedef g4i __attribute__((address_space(3))) lds4i;
__device__ __forceinline__ glb4i* to_glb(const void* p) {
  return (glb4i*)(__UINTPTR_TYPE__)p;
}
__device__ __forceinline__ lds4i* to_lds(void* p) {
  // generic LDS pointer: low 32 bits are the LDS byte offset (ISA §10.2)
  return (lds4i*)(__UINT32_TYPE__)(__UINTPTR_TYPE__)p;
}
#endif

// XOR swizzle at 16B (4-float) granularity: conflict-free column reads from
// row-major [64][64] LDS tiles without padding (async B128 needs 16B blocks).
__device__ __forceinline__ int swz(int row, int col) {
  return (col & 3) | ((((col >> 2) ^ row) & 15) << 2);
}

// =====================================================================
// Kernel 1: qkv = x @ w_qkv^T, scattered into q/k/v with [B,H,S,hd] layout
// C[M=4096, N=3072], K=1024. 64x64 tile per block, 4 waves, 2x2 WMMA/wave.
// =====================================================================
__global__ __launch_bounds__(128)
void qkv_gemm(const float* __restrict__ X, const float* __restrict__ W,
              float* __restrict__ qb, float* __restrict__ kb, float* __restrict__ vb) {
  __shared__ float As[64][17];
  __shared__ float Bs[64][17];
  const int tid  = threadIdx.x;
  const int lane = tid & 31;
  const int wave = tid >> 5;
  const int lm = lane & 15, lh = lane >> 4;
  const int wm = wave >> 1, wn = wave & 1;
  const int gm0 = blockIdx.y * 64;
  const int gn0 = blockIdx.x * 64;

  v8f acc[2][2] = {};

#pragma unroll 1
  for (int k0 = 0; k0 < DD; k0 += 16) {
    __syncthreads();
    for (int i = tid; i < 256; i += 128) {
      int row = i >> 2, c4 = (i & 3) * 4;
      v4f a = *(const v4f*)(X + (size_t)(gm0 + row) * DD + k0 + c4);
      As[row][c4+0] = a.x; As[row][c4+1] = a.y; As[row][c4+2] = a.z; As[row][c4+3] = a.w;
      v4f bb = *(const v4f*)(W + (size_t)(gn0 + row) * DD + k0 + c4);
      Bs[row][c4+0] = bb.x; Bs[row][c4+1] = bb.y; Bs[row][c4+2] = bb.z; Bs[row][c4+3] = bb.w;
    }
    __syncthreads();
#pragma unroll
    for (int kk = 0; kk < 4; ++kk) {
      const int kidx = kk * 4 + lh * 2;    // A/B frag: K = 2*(lane/16) + j
      v2f a0, a1, b0, b1;
      a0.x = As[wm*32      + lm][kidx]; a0.y = As[wm*32      + lm][kidx+1];
      a1.x = As[wm*32 + 16 + lm][kidx]; a1.y = As[wm*32 + 16 + lm][kidx+1];
      b0.x = Bs[wn*32      + lm][kidx]; b0.y = Bs[wn*32      + lm][kidx+1];
      b1.x = Bs[wn*32 + 16 + lm][kidx]; b1.y = Bs[wn*32 + 16 + lm][kidx+1];
      acc[0][0] = wmma4(a0, b0, acc[0][0]);
      acc[0][1] = wmma4(a0, b1, acc[0][1]);
      acc[1][0] = wmma4(a1, b0, acc[1][0]);
      acc[1][1] = wmma4(a1, b1, acc[1][1]);
    }
  }

  // scatter epilogue: split qkv into q/k/v with [B,H,S,hd] layout
#pragma unroll
  for (int i = 0; i < 2; ++i)
#pragma unroll
    for (int j = 0; j < 2; ++j)
#pragma unroll
      for (int r = 0; r < 8; ++r) {
        int m = gm0 + wm*32 + i*16 + r + lh*8;   // C row: M = v + 8*(lane/16)
        int n = gn0 + wn*32 + j*16 + lm;         // C col: N = lane%16
        float val = acc[i][j][r];
        int b = m / SS, s = m % SS;
        int part = n / DD, rem = n % DD;
        int h = rem / HD, d = rem % HD;
        float* dst = (part == 0) ? qb : ((part == 1) ? kb : vb);
        dst[(((size_t)b * HH + h) * SS + s) * HD + d] = val;
      }
}

// =====================================================================
// Kernel 2: block router (only batch 0 matters in the reference).
// =====================================================================
__global__ __launch_bounds__(128)
void router_kernel(const float* __restrict__ kb, const float* __restrict__ w_r1,
                   const float* __restrict__ b_r1, const float* __restrict__ w_r2,
                   const float* __restrict__ b_r2, int* __restrict__ keep) {
  __shared__ float mean[DD];
  __shared__ float red[128];
  const int j = blockIdx.x, tid = threadIdx.x;

  for (int c = tid; c < DD; c += 128) {
    int h = c >> 6, d = c & 63;
    const float* base = kb + ((size_t)h * SS) * HD + d;   // batch 0
    float acc = 0.f;
    const int s0 = j * 64;
    for (int s = 0; s < 64; ++s) acc += base[(size_t)(s0 + s) * HD];
    mean[c] = acc * (1.0f / 64.0f);
  }
  __syncthreads();

  float z = b_r1[tid];
  const float* wr = w_r1 + (size_t)tid * DD;
  for (int k = 0; k < DD; ++k) z += mean[k] * wr[k];
  z = fmaxf(z, 0.f);
  red[tid] = z * w_r2[tid];
  __syncthreads();
  for (int off = 64; off > 0; off >>= 1) {
    if (tid < off) red[tid] += red[tid + off];
    __syncthreads();
  }
  if (tid == 0) keep[j] = (red[0] + b_r2[0]) > 0.f ? 1 : 0;   // sigmoid(x)>0.5 <=> x>0
}

// =====================================================================
// Kernel 3: flash attention with block-router + causal masks.
// grid (S/64, H, B); 4 waves; wave w owns query rows [w*16, w*16+16).
// K/V tiles staged LDS-direct via async loads (swizzled), P relayout via LDS.
// =====================================================================
__global__ __launch_bounds__(128)
void flash_attn(const float* __restrict__ qb, const float* __restrict__ kb,
                const float* __restrict__ vb, const int* __restrict__ keep,
                float* __restrict__ ao) {
  __shared__ float Kt[64][64];   // XOR-swizzled (16B granularity)
  __shared__ float Vt[64][64];   // XOR-swizzled
  __shared__ float Ps[64][65];   // padded, normal ds traffic

  const int qblk = blockIdx.x, h = blockIdx.y, b = blockIdx.z;
  const int tid = threadIdx.x, lane = tid & 31, wave = tid >> 5;
  const int lm = lane & 15, lh = lane >> 4;

  const size_t headBase = (((size_t)b * HH + h) * SS) * HD;
  const int qrow0 = qblk * 64 + wave * 16;

  // resident Q A-fragments: aQ[t] covers d = 4t..4t+3
  v2f aQ[16];
  {
    const float* qrow = qb + headBase + (size_t)(qrow0 + lm) * HD;
#pragma unroll
    for (int t = 0; t < 16; ++t) {
      int d = t * 4 + lh * 2;
      aQ[t].x = qrow[d]; aQ[t].y = qrow[d + 1];
    }
  }

  v8f accO[4] = {};
  float mrow[8], lrow[8];
#pragma unroll
  for (int r = 0; r < 8; ++r) { mrow[r] = -INFINITY; lrow[r] = 0.f; }

  const bool keepQ = keep[qblk] != 0;
  const float scale = 0.03125f;  // 1/sqrt(d_model=1024)

#pragma unroll 1
  for (int kbi = 0; kbi <= qblk; ++kbi) {
    const bool keepK = keep[kbi] != 0;
    if (keepQ && !keepK) continue;         // exp(-1e9 - m) == 0 in fp32: exact skip
    const bool masked = !(keepQ && keepK); // all-(-1e9) path when !keepQ

    __syncthreads();                       // previous tile fully consumed
    const float* Kg = kb + headBase + (size_t)kbi * 64 * HD;
    const float* Vg = vb + headBase + (size_t)kbi * 64 * HD;

#ifdef USE_ASYNC_LDS
    // 64x64 tile = 1024 16B blocks; 8 async B128 issues per thread per tile
    for (int i = tid; i < 1024; i += 128) {
      int row = i >> 4, g = i & 15;
      int gp = (g ^ row) & 15;
      __builtin_amdgcn_global_load_async_to_lds_b128(
          to_glb(Kg + (size_t)row * HD + g * 4), to_lds(&Kt[row][gp * 4]), 0, 0);
      __builtin_amdgcn_global_load_async_to_lds_b128(
          to_glb(Vg + (size_t)row * HD + g * 4), to_lds(&Vt[row][gp * 4]), 0, 0);
    }
#else
    for (int i = tid; i < 1024; i += 128) {
      int row = i >> 4, g = i & 15;
      int c4 = ((g ^ row) & 15) * 4;
      v4f kv = *(const v4f*)(Kg + (size_t)row * HD + g * 4);
      Kt[row][c4+0] = kv.x; Kt[row][c4+1] = kv.y; Kt[row][c4+2] = kv.z; Kt[row][c4+3] = kv.w;
      v4f vv = *(const v4f*)(Vg + (size_t)row * HD + g * 4);
      Vt[row][c4+0] = vv.x; Vt[row][c4+1] = vv.y; Vt[row][c4+2] = vv.z; Vt[row][c4+3] = vv.w;
    }
#endif

    // prefetch next k/v tile into cache while this one is consumed
    if (kbi < qblk) {
      const float* nK = Kg + (size_t)64 * HD;
      const float* nV = Vg + (size_t)64 * HD;
      __builtin_prefetch(nK + (size_t)tid * 32, 0, 1);
      __builtin_prefetch(nV + (size_t)tid * 32, 0, 1);
    }

#ifdef USE_ASYNC_LDS
    __builtin_amdgcn_s_wait_asynccnt(0);
#endif
    __syncthreads();

    // S = Q @ K^T for this wave's 16 rows x 64 key cols
    v8f accS[4] = {};
#pragma unroll
    for (int n = 0; n < 4; ++n) {
      const int krow = n * 16 + lm;
#pragma unroll
      for (int t = 0; t < 16; ++t) {
        int d = t * 4 + lh * 2;            // d, d+1 stay in one 16B group
        int c = swz(krow, d);
        v2f bk; bk.x = Kt[krow][c]; bk.y = Kt[krow][c + 1];
        accS[n] = wmma4(aQ[t], bk, accS[n]);
      }
    }

    // scale + block mask + causal mask (element: row=r+8*lh, col=n*16+lm)
    float sv[4][8];
#pragma unroll
    for (int n = 0; n < 4; ++n)
#pragma unroll
      for (int r = 0; r < 8; ++r) {
        float x = accS[n][r] * scale;
        if (masked) x = NEG_BIG;
        int col = kbi * 64 + n * 16 + lm;
        int row = qblk * 64 + wave * 16 + r + lh * 8;
        if (col > row) x = -INFINITY;
        sv[n][r] = x;
      }

    // online softmax; rows 0-7 in lanes 0-15, rows 8-15 in lanes 16-31:
    // shfl_xor masks 1..8 reduce exactly over each row's 16 column-lanes.
#pragma unroll
    for (int r = 0; r < 8; ++r) {
      float mx = fmaxf(fmaxf(sv[0][r], sv[1][r]), fmaxf(sv[2][r], sv[3][r]));
#pragma unroll
      for (int off = 1; off < 16; off <<= 1) mx = fmaxf(mx, __shfl_xor(mx, off, 32));
      float mnew  = fmaxf(mrow[r], mx);          // finite once any block processed
      float alpha = __expf(mrow[r] - mnew);      // exp(-inf)=0 on first pass
      float rs = 0.f;
#pragma unroll
      for (int n = 0; n < 4; ++n) {
        float p = __expf(sv[n][r] - mnew);
        sv[n][r] = p;
        rs += p;
      }
#pragma unroll
      for (int off = 1; off < 16; off <<= 1) rs += __shfl_xor(rs, off, 32);
      lrow[r] = lrow[r] * alpha + rs;
      mrow[r] = mnew;
#pragma unroll
      for (int n = 0; n < 4; ++n) accO[n][r] *= alpha;
    }

    // relayout P (C-layout -> A-layout) through LDS; rows are wave-private
#pragma unroll
    for (int n = 0; n < 4; ++n)
#pragma unroll
      for (int r = 0; r < 8; ++r)
        Ps[wave * 16 + r + lh * 8][n * 16 + lm] = sv[n][r];
    __syncthreads();

    v2f aP[16];
#pragma unroll
    for (int t = 0; t < 16; ++t) {
      int kk = t * 4 + lh * 2;
      aP[t].x = Ps[wave * 16 + lm][kk];
      aP[t].y = Ps[wave * 16 + lm][kk + 1];
    }
    // O += P @ V
#pragma unroll
    for (int n2 = 0; n2 < 4; ++n2) {
      const int vc = n2 * 16 + lm;
#pragma unroll
      for (int t = 0; t < 16; ++t) {
        int kk = t * 4 + lh * 2;
        v2f bv;
        bv.x = Vt[kk    ][swz(kk,     vc)];
        bv.y = Vt[kk + 1][swz(kk + 1, vc)];
        accO[n2] = wmma4(aP[t], bv, accO[n2]);
      }
    }
  }

  // normalize and store to [B,S,D] (D index = h*64 + d) for the out-proj GEMM
#pragma unroll
  for (int n2 = 0; n2 < 4; ++n2)
#pragma unroll
    for (int r = 0; r < 8; ++r) {
      int srow = qrow0 + r + lh * 8;
      int d = n2 * 16 + lm;
      float o = accO[n2][r] / lrow[r];
      ao[((size_t)b * SS + srow) * DD + h * HD + d] = o;
    }
}

// =====================================================================
// Kernel 4: out = ao @ w_out^T + b_out.  C[4096,1024], K=1024.
// =====================================================================
__global__ __launch_bounds__(128)
void out_gemm(const float* __restrict__ A, const float* __restrict__ W,
              const float* __restrict__ bias, float* __restrict__ C) {
  __shared__ float As[64][17];
  __shared__ float Bs[64][17];
  const int tid  = threadIdx.x;
  const int lane = tid & 31;
  const int wave = tid >> 5;
  const int lm = lane & 15, lh = lane >> 4;
  const int wm = wave >> 1, wn = wave & 1;
  const int gm0 = blockIdx.y * 64;
  const int gn0 = blockIdx.x * 64;

  v8f acc[2][2] = {};

#pragma unroll 1
  for (int k0 = 0; k0 < DD; k0 += 16) {
    __syncthreads();
    for (int i = tid; i < 256; i += 128) {
      int row = i >> 2, c4 = (i & 3) * 4;
      v4f a = *(const v4f*)(A + (size_t)(gm0 + row) * DD + k0 + c4);
      As[row][c4+0] = a.x; As[row][c4+1] = a.y; As[row][c4+2] = a.z; As[row][c4+3] = a.w;
      v4f bb = *(const v4f*)(W + (size_t)(gn0 + row) * DD + k0 + c4);
      Bs[row][c4+0] = bb.x; Bs[row][c4+1] = bb.y; Bs[row][c4+2] = bb.z; Bs[row][c4+3] = bb.w;
    }
    __syncthreads();
#pragma unroll
    for (int kk = 0; kk < 4; ++kk) {
      const int kidx = kk * 4 + lh * 2;
      v2f a0, a1, b0, b1;
      a0.x = As[wm*32      + lm][kidx]; a0.y = As[wm*32      + lm][kidx+1];
      a1.x = As[wm*32 + 16 + lm][kidx]; a1.y = As[wm*32 + 16 + lm][kidx+1];
      b0.x = Bs[wn*32      + lm][kidx]; b0.y = Bs[wn*32      + lm][kidx+1];
      b1.x = Bs[wn*32 + 16 + lm][kidx]; b1.y = Bs[wn*32 + 16 + lm][kidx+1];
      acc[0][0] = wmma4(a0, b0, acc[0][0]);
      acc[0][1] = wmma4(a0, b1, acc[0][1]);
      acc[1][0] = wmma4(a1, b0, acc[1][0]);
      acc[1][1] = wmma4(a1, b1, acc[1][1]);
    }
  }

#pragma unroll
  for (int i = 0; i < 2; ++i)
#pragma unroll
    for (int j = 0; j < 2; ++j)
#pragma unroll
      for (int r = 0; r < 8; ++r) {
        int m = gm0 + wm*32 + i*16 + r + lh*8;
        int n = gn0 + wn*32 + j*16 + lm;
        C[(size_t)m * DD + n] = acc[i][j][r] + bias[n];
      }
}

// =====================================================================
extern "C" void kernel_launch(void* const* d_in, const int* in_sizes, int n_in,
                              void* d_out, int out_size, void* d_ws, size_t ws_size,
                              hipStream_t stream) {
  (void)in_sizes; (void)n_in; (void)out_size; (void)ws_size;
  const float* x     = (const float*)d_in[0];
  const float* w_qkv = (const float*)d_in[1];
  const float* w_r1  = (const float*)d_in[2];
  const float* b_r1  = (const float*)d_in[3];
  const float* w_r2  = (const float*)d_in[4];
  const float* b_r2  = (const float*)d_in[5];
  const float* w_out = (const float*)d_in[6];
  const float* b_out = (const float*)d_in[7];
  float* out = (float*)d_out;

  const size_t per = (size_t)2 * HH * SS * HD;   // 4,194,304 floats per q/k/v/ao
  float* ws   = (float*)d_ws;
  float* qb   = ws;
  float* kbuf = qb + per;
  float* vbuf = kbuf + per;
  float* ao   = vbuf + per;
  int*   keep = (int*)(ao + per);

  qkv_gemm<<<dim3(3 * DD / 64, (2 * SS) / 64), 128, 0, stream>>>(x, w_qkv, qb, kbuf, vbuf);
  router_kernel<<<dim3(NBLK), 128, 0, stream>>>(kbuf, w_r1, b_r1, w_r2, b_r2, keep);
  flash_attn<<<dim3(SS / 64, HH, 2), 128, 0, stream>>>(qb, kbuf, vbuf, keep, ao);
  out_gemm<<<dim3(DD / 64, (2 * SS) / 64), 128, 0, stream>>>(ao, w_out, b_out, out);
}